// select_class_max_79182017069248
// MI455X (gfx1250) — compile-verified
//
#include <hip/hip_runtime.h>

#define B_ 8
#define N_ 4096
#define F_ 1024
#define C_ 64
#define KC 128     // K chunk staged in LDS
#define KCP 132    // padded LDS stride (floats): 16B-aligned rows, conflict-free b64 reads

typedef __attribute__((ext_vector_type(2))) float v2f;
typedef __attribute__((ext_vector_type(4))) float v4f;
typedef __attribute__((ext_vector_type(8))) float v8f;

// Monotone order-preserving f32 -> u32 (larger float => larger uint)
__device__ __forceinline__ unsigned int sortable(float f) {
  unsigned int u = __float_as_uint(f);
  return (u & 0x80000000u) ? ~u : (u | 0x80000000u);
}

__global__ void init_keys_kernel(unsigned long long* keys, int n) {
  int i = blockIdx.x * blockDim.x + threadIdx.x;
  if (i < n) keys[i] = 0ull;
}

// scores[b] = x[b] (4096x1024) * W^T (1024x64), per-class argmax over rows.
// Each block: 4 waves x 16 rows = 64 rows of batch blockIdx.y.
// Each wave: 4 accumulators (16x16 f32) covering all 64 classes.
__global__ void __launch_bounds__(128) score_argmax_kernel(
    const float* __restrict__ x, const float* __restrict__ W,
    unsigned long long* __restrict__ keys) {
  __shared__ float lds_w[C_ * KCP];

  const int tid  = threadIdx.x;
  const int wave = tid >> 5;
  const int lane = tid & 31;
  const int half = lane >> 4;   // 0 -> K pair {0,1}; 1 -> K pair {2,3}
  const int l16  = lane & 15;   // row (A) / class-in-tile (B,C,D)
  const int b    = blockIdx.y;
  const int rowBase = blockIdx.x * 64 + wave * 16;

  v8f acc[4];
  #pragma unroll
  for (int t = 0; t < 4; ++t)
    #pragma unroll
    for (int r = 0; r < 8; ++r) acc[t][r] = 0.0f;

  // A fragment addressing: lanes 0..15 and 16..31 both map rows rowBase..+15,
  // halves differ in the K pair (32-bit 16x4 A layout).
  const float* aRow =
      x + ((size_t)b * N_ + (size_t)(rowBase + l16)) * F_ + 2 * half;

  for (int kko = 0; kko < F_; kko += KC) {
    __syncthreads();
    // Stage W[0:64][kko : kko+KC] into LDS with coalesced b128 loads.
    for (int j = tid; j < C_ * (KC / 4); j += 128) {
      int c  = j >> 5;   // 32 float4 per class row chunk
      int k4 = j & 31;
      v4f v = *reinterpret_cast<const v4f*>(&W[(size_t)c * F_ + kko + k4 * 4]);
      *reinterpret_cast<v4f*>(&lds_w[c * KCP + k4 * 4]) = v;
    }
    __syncthreads();

    #pragma unroll 4
    for (int kk = 0; kk < KC; kk += 4) {
      v2f a = *reinterpret_cast<const v2f*>(&aRow[kko + kk]);
      #pragma unroll
      for (int t = 0; t < 4; ++t) {
        // B fragment (4x16, K x class): B[k][c] = W[c][k]
        v2f bb = *reinterpret_cast<const v2f*>(
            &lds_w[(t * 16 + l16) * KCP + kk + 2 * half]);
        acc[t] = __builtin_amdgcn_wmma_f32_16x16x4_f32(
            false, a, false, bb, (short)0, acc[t], false, false);
      }
    }
  }

  // D layout: lane L, reg r  ->  scores[row = rowBase + 8*(L>>4) + r]
  //                                    [class = t*16 + (L&15)]
  // Reduce 16 rows per class, then merge globally via packed u64 atomic max.
  #pragma unroll
  for (int t = 0; t < 4; ++t) {
    unsigned long long best = 0ull;
    #pragma unroll
    for (int r = 0; r < 8; ++r) {
      unsigned int row = (unsigned int)(rowBase + half * 8 + r);
      unsigned long long key =
          ((unsigned long long)sortable(acc[t][r]) << 32) |
          (unsigned long long)(~row);   // ties -> smallest row wins (first occurrence)
      best = key > best ? key : best;
    }
    unsigned long long other = __shfl_xor(best, 16, 32);
    best = other > best ? other : best;
    if (lane < 16)
      atomicMax(&keys[(size_t)b * C_ + t * 16 + l16], best);
  }
}

// One block per (tensor, b, c): copy the winning 1024-float row.
__global__ void __launch_bounds__(256) gather_kernel(
    const float* __restrict__ x1, const float* __restrict__ x2,
    const unsigned long long* __restrict__ keys, float* __restrict__ out) {
  int blk = blockIdx.x;          // o*512 + b*64 + c
  int o  = blk >> 9;
  int bc = blk & 511;
  int b  = bc >> 6;
  int c  = bc & 63;
  unsigned long long key = keys[(size_t)o * B_ * C_ + (size_t)b * C_ + c];
  unsigned int row = ~(unsigned int)(key & 0xFFFFFFFFull);
  const float* src = (o ? x2 : x1) + ((size_t)b * N_ + row) * F_;
  float* dst = out + (((size_t)o * B_ * C_) + (size_t)b * C_ + c) * F_;
  int t = threadIdx.x;           // 256 threads x float4 = 1024 floats
  v4f v = *reinterpret_cast<const v4f*>(&src[t * 4]);
  *reinterpret_cast<v4f*>(&dst[t * 4]) = v;
}

extern "C" void kernel_launch(void* const* d_in, const int* in_sizes, int n_in,
                              void* d_out, int out_size, void* d_ws, size_t ws_size,
                              hipStream_t stream) {
  (void)in_sizes; (void)n_in; (void)out_size; (void)ws_size;
  const float* x1 = (const float*)d_in[0];
  const float* x2 = (const float*)d_in[1];
  const float* W  = (const float*)d_in[2];
  // d_in[3] (bias) is per-class constant: cannot change within-class argmax -> unused.

  unsigned long long* keys = (unsigned long long*)d_ws;  // [2][B][C] u64
  const int nkeys = 2 * B_ * C_;

  init_keys_kernel<<<(nkeys + 255) / 256, 256, 0, stream>>>(keys, nkeys);

  dim3 grid(N_ / 64, B_);
  score_argmax_kernel<<<grid, 128, 0, stream>>>(x1, W, keys);
  score_argmax_kernel<<<grid, 128, 0, stream>>>(x2, W, keys + B_ * C_);

  gather_kernel<<<2 * B_ * C_, 256, 0, stream>>>(x1, x2, keys, (float*)d_out);
}